// MoE_3023656976530
// MI455X (gfx1250) — compile-verified
//
#include <hip/hip_runtime.h>

// MoE top-1 routed conv-expert, fused for MI455X (gfx1250, wave32).
// B=2048 tokens, D=4096, P=64 patches, K=64, E=8 experts, 2C=128 channels.
// One wave per token: router -> argmax -> selected-expert GEMM via
// V_WMMA_F32_16X16X4_F32 -> cube/reduce -> softmax.

typedef __attribute__((ext_vector_type(2))) float v2f;
typedef __attribute__((ext_vector_type(8))) float v8f;

#define B_  2048
#define D_  4096
#define P_  64
#define E_  8
#define K_  64
#define C2_ 128

// d_out layout (fp32): [0,4096) out[B,2] | [4096,20480) select0[B,E] | [20480] loss

__global__ __launch_bounds__(256) void moe_fused_kernel(
    const float* __restrict__ x,         // [B, D]
    const float* __restrict__ router_w,  // [E, K]
    const float* __restrict__ expert_w,  // [E, 2C, K]
    const float* __restrict__ expert_b,  // [E, 2C]
    float* __restrict__ out,
    float* __restrict__ ws_sel,          // [B, E] scratch
    int* __restrict__ ws_idx)            // [B] scratch
{
  __shared__ float sk[8][K_];
  __shared__ float sel[8][E_];
  __shared__ float gateS[8];
  __shared__ int   idxS[8];

  const int lane = threadIdx.x & 31;
  const int w    = threadIdx.x >> 5;           // wave id in block (8 waves)
  const int b    = blockIdx.x * 8 + w;         // one token per wave
  const float* __restrict__ xr = x + b * D_;

  // ---------- Router: s_k[k] = sum_p x[b, p*K + k]; select[e] = <s_k, rw[e]>
  float s0 = 0.f, s1 = 0.f;
  for (int p = 0; p < P_; ++p) {
    s0 += xr[p * K_ + lane];
    s1 += xr[p * K_ + lane + 32];
  }
  sk[w][lane]      = s0;
  sk[w][lane + 32] = s1;
  __syncthreads();

  if (lane < E_) {
    float acc = 0.f;
    const float* __restrict__ rw = router_w + lane * K_;
#pragma unroll
    for (int k = 0; k < K_; ++k) acc += sk[w][k] * rw[k];
    sel[w][lane] = acc;
    ws_sel[b * E_ + lane] = acc;               // for deterministic loss reduce
  }
  __syncthreads();

  if (lane == 0) {
    float mx = sel[w][0];
    int   im = 0;
    for (int e = 1; e < E_; ++e) {             // strict '>' == first-occurrence argmax
      float v = sel[w][e];
      if (v > mx) { mx = v; im = e; }
    }
    gateS[w] = mx;
    idxS[w]  = im;
    ws_idx[b] = im;
  }
  __syncthreads();

  const float gate = gateS[w];
  const int   eSel = idxS[w];
  if (lane < E_)
    out[4096 + b * E_ + lane] = (lane == eSel && gate != 0.f) ? 1.f : 0.f;

  // ---------- Selected-expert GEMM: D[p,c] = sum_k x[b,p,k] * W[eSel,c,k]
  // M=p (A from x, row-major [p][k]); N=c (B from W, row-major [c][k] == B^T).
  // Both fragments are direct float2 global loads per lane.
  const float dmask = (gate != 0.f) ? 1.f : 0.f;
  const float* __restrict__ Wm   = expert_w + eSel * (C2_ * K_);
  const float* __restrict__ bias = expert_b + eSel * C2_;
  const int lh   = lane & 15;
  const int kofs = (lane >> 4) << 1;           // lanes 16-31 hold K+2, K+3

  float sumA = 0.f, sumB = 0.f;                // cube-sums for c<64 / c>=64
  for (int mt = 0; mt < 4; ++mt) {             // p tiles of 16
    const float* __restrict__ ap = xr + (mt * 16 + lh) * K_ + kofs;
    v2f a[16];
#pragma unroll
    for (int kk = 0; kk < 16; ++kk)
      a[kk] = *(const v2f*)(ap + kk * 4);      // A frag reused over 8 c-tiles

    for (int nt = 0; nt < 8; ++nt) {           // c tiles of 16
      const float bb = bias[nt * 16 + lh];     // c = c0 + (lane&15): const over v
      const float* __restrict__ bp = Wm + (nt * 16 + lh) * K_ + kofs;
      v8f acc = {0.f, 0.f, 0.f, 0.f, 0.f, 0.f, 0.f, 0.f};
#pragma unroll
      for (int kk = 0; kk < 16; ++kk) {
        v2f bf = *(const v2f*)(bp + kk * 4);
        acc = __builtin_amdgcn_wmma_f32_16x16x4_f32(
            /*neg_a=*/false, a[kk], /*neg_b=*/false, bf,
            /*c_mod=*/(short)0, acc, /*reuse_a=*/false, /*reuse_b=*/false);
      }
      float loc = 0.f;
#pragma unroll
      for (int v = 0; v < 8; ++v) {
        float t = acc[v] * dmask + bb;         // z*dispatch + bias
        loc += t * t * t;                      // z^3, summed over p and c-half
      }
      if (nt < 4) sumA += loc; else sumB += loc;
    }
  }

  // wave32 tree reduction
#pragma unroll
  for (int off = 16; off > 0; off >>= 1) {
    sumA += __shfl_xor(sumA, off, 32);
    sumB += __shfl_xor(sumB, off, 32);
  }

  if (lane == 0) {
    float o0 = gate * sumA, o1 = gate * sumB;  // gate==0 -> softmax([0,0])=[.5,.5]
    float m  = fmaxf(o0, o1);
    float e0 = expf(o0 - m), e1 = expf(o1 - m);
    float inv = 1.f / (e0 + e1);
    out[b * 2 + 0] = e0 * inv;
    out[b * 2 + 1] = e1 * inv;
  }
}

// Deterministic fixed-order reduction for the aux load-balance loss.
__global__ __launch_bounds__(256) void moe_loss_kernel(
    const float* __restrict__ ws_sel, const int* __restrict__ ws_idx,
    float* __restrict__ out)
{
  __shared__ float S[256];
  __shared__ float Cn[256];
  const int e = threadIdx.x & 7;
  const int g = threadIdx.x >> 3;              // 32 row-groups per expert
  float s = 0.f, c = 0.f;
  for (int b = g; b < B_; b += 32) {
    s += ws_sel[b * E_ + e];
    c += (ws_idx[b] == e) ? 1.f : 0.f;
  }
  S[threadIdx.x]  = s;
  Cn[threadIdx.x] = c;
  __syncthreads();
  for (int st = 16; st > 0; st >>= 1) {
    if (g < st) {
      S[threadIdx.x]  += S[threadIdx.x + st * 8];
      Cn[threadIdx.x] += Cn[threadIdx.x + st * 8];
    }
    __syncthreads();
  }
  if (threadIdx.x == 0) {
    const float invB = 1.f / 2048.f;
    float loss = 0.f;
    for (int ee = 0; ee < E_; ++ee)
      loss += (S[ee] * invB) * (Cn[ee] * invB);
    out[20480] = loss * 8.f;                   // mean over E, times E^2
  }
}

extern "C" void kernel_launch(void* const* d_in, const int* in_sizes, int n_in,
                              void* d_out, int out_size, void* d_ws, size_t ws_size,
                              hipStream_t stream) {
  (void)in_sizes; (void)n_in; (void)out_size; (void)ws_size;
  const float* x        = (const float*)d_in[0];
  const float* router_w = (const float*)d_in[1];
  const float* expert_w = (const float*)d_in[2];
  const float* expert_b = (const float*)d_in[3];
  float* out = (float*)d_out;
  float* ws_sel = (float*)d_ws;                       // B*E floats = 64 KB
  int*   ws_idx = (int*)((char*)d_ws + B_ * E_ * 4);  // B ints = 8 KB

  moe_fused_kernel<<<B_ / 8, 256, 0, stream>>>(x, router_w, expert_w, expert_b,
                                               out, ws_sel, ws_idx);
  moe_loss_kernel<<<1, 256, 0, stream>>>(ws_sel, ws_idx, out);
}